// CRGCN_1288490189295
// MI455X (gfx1250) — compile-verified
//
#include <hip/hip_runtime.h>
#include <math.h>

#define NN      150002      // total nodes (N_USERS+1 + N_ITEMS+1)
#define NU      50001       // user rows
#define NI      100001      // item rows
#define DD      64
#define NE      1000000
#define BATCHSZ 4096

typedef float v2f __attribute__((ext_vector_type(2)));
typedef float v8f __attribute__((ext_vector_type(8)));

// ---------------------------------------------------------------- utilities
__global__ void k_init_total(const float* __restrict__ uemb,
                             const float* __restrict__ iemb,
                             float* __restrict__ total) {
  size_t i = (size_t)blockIdx.x * blockDim.x + threadIdx.x;
  if (i >= (size_t)NN * DD) return;
  size_t row = i >> 6, col = i & 63;
  total[i] = (row < NU) ? uemb[row * DD + col] : iemb[(row - NU) * DD + col];
}

__global__ void k_zero(float* __restrict__ p, int n) {
  int i = blockIdx.x * blockDim.x + threadIdx.x;
  if (i < n) p[i] = 0.f;
}

__global__ void k_copy(const float* __restrict__ s, float* __restrict__ d, size_t n) {
  size_t i = (size_t)blockIdx.x * blockDim.x + threadIdx.x;
  if (i < n) d[i] = s[i];
}

__global__ void k_sumsq(const float* __restrict__ x, size_t n, float* __restrict__ acc) {
  size_t i0 = (size_t)blockIdx.x * blockDim.x + threadIdx.x;
  size_t stride = (size_t)gridDim.x * blockDim.x;
  float s = 0.f;
  for (size_t i = i0; i < n; i += stride) { float v = x[i]; s += v * v; }
  #pragma unroll
  for (int m = 16; m > 0; m >>= 1) s += __shfl_xor(s, m, 32);
  if ((threadIdx.x & 31) == 0) atomicAdd(acc, s);
}

// ---------------------------------------------------------------- graph prep
__global__ void k_degree(const int* __restrict__ dst, float* __restrict__ deg) {
  int e = blockIdx.x * blockDim.x + threadIdx.x;
  if (e < NE) atomicAdd(&deg[dst[e]], 1.0f);
}

__global__ void k_dinv(const float* __restrict__ deg, float* __restrict__ dinv) {
  int i = blockIdx.x * blockDim.x + threadIdx.x;
  if (i >= NN) return;
  float dg = deg[i];
  dinv[i] = (dg > 0.f) ? rsqrtf(fmaxf(dg, 1.f)) : 0.f;
}

// ---------------------------------------------------------------- WMMA GEMM
// H[nrows x 64] = X[nrows x 64] @ W[64 x 64], fp32 via v_wmma_f32_16x16x4_f32.
// One wave handles a 16-row strip: 4 accumulators of 16x16 covering N=64.
__global__ void k_gemm_wmma(const float* __restrict__ X, const float* __restrict__ W,
                            float* __restrict__ H, int nrows) {
  const int lane  = threadIdx.x & 31;
  const int wave  = threadIdx.x >> 5;
  const int strip = blockIdx.x * (blockDim.x >> 5) + wave;
  const int row0  = strip << 4;
  if (row0 >= nrows) return;                 // wave-uniform exit

  const int m     = lane & 15;
  const int khalf = (lane >> 4) << 1;        // lanes 0-15 -> K 0,1 ; lanes 16-31 -> K 2,3
  int rowA = row0 + m;
  if (rowA >= nrows) rowA = nrows - 1;       // clamp (tail garbage rows never stored)
  const float* xr = X + (size_t)rowA * DD;
  const int n = lane & 15;

  v8f acc[4] = {};
  #pragma unroll
  for (int k0 = 0; k0 < DD; k0 += 4) {
    v2f a;                                    // A: 16x4, row m, cols k0+khalf(+1)
    a.x = xr[k0 + khalf + 0];
    a.y = xr[k0 + khalf + 1];
    #pragma unroll
    for (int nt = 0; nt < 4; ++nt) {
      v2f b;                                  // B: 4x16, VGPR j -> K = k0 + j + khalf
      b.x = W[(k0 + khalf + 0) * DD + nt * 16 + n];
      b.y = W[(k0 + khalf + 1) * DD + nt * 16 + n];
      acc[nt] = __builtin_amdgcn_wmma_f32_16x16x4_f32(
          false, a, false, b, (short)0, acc[nt], false, false);
    }
  }

  const int mbase = (lane >> 4) << 3;         // C layout: VGPR j -> M = j + 8*(lane>=16)
  #pragma unroll
  for (int nt = 0; nt < 4; ++nt) {
    #pragma unroll
    for (int j = 0; j < 8; ++j) {
      int r = row0 + mbase + j;
      if (r < nrows) H[(size_t)r * DD + nt * 16 + n] = acc[nt][j];
    }
  }
}

// ---------------------------------------------------------------- scatter-add
__global__ void k_bias_init(float* __restrict__ P, const float* __restrict__ b) {
  size_t i = (size_t)blockIdx.x * blockDim.x + threadIdx.x;
  if (i < (size_t)NN * DD) P[i] = b[i & 63];
}

// P[dst] += H[src] * dinv[src]*dinv[dst]; 32 lanes per edge, float2 per lane.
__global__ void k_scatter(const int* __restrict__ src, const int* __restrict__ dst,
                          const float* __restrict__ dinv, const float* __restrict__ H,
                          float* __restrict__ P) {
  long long t = (long long)blockIdx.x * blockDim.x + threadIdx.x;
  int e    = (int)(t >> 5);
  int lane = (int)(t & 31);
  if (e >= NE) return;
  int s = src[e], d = dst[e];
  float en = dinv[s] * dinv[d];
  if (en == 0.f) return;
  float2 v = ((const float2*)(H + (size_t)s * DD))[lane];
  float* pd = P + (size_t)d * DD + lane * 2;
  atomicAdd(pd + 0, v.x * en);
  atomicAdd(pd + 1, v.y * en);
}

// ---------------------------------------------------------------- combine
// total += (x / max(||x||,1e-12)) * layer_weight ; one wave per node row.
__global__ void k_combine(const float* __restrict__ X, const float* __restrict__ LW,
                          float* __restrict__ total) {
  int t    = blockIdx.x * blockDim.x + threadIdx.x;
  int node = t >> 5;
  int lane = t & 31;
  if (node >= NN) return;
  float2 v = ((const float2*)(X + (size_t)node * DD))[lane];
  float s = v.x * v.x + v.y * v.y;
  #pragma unroll
  for (int m = 16; m > 0; m >>= 1) s += __shfl_xor(s, m, 32);
  float inv = 1.f / fmaxf(sqrtf(s), 1e-12f);
  size_t base = (size_t)node * DD + lane * 2;
  total[base + 0] += v.x * inv * LW[base + 0];
  total[base + 1] += v.y * inv * LW[base + 1];
}

// ---------------------------------------------------------------- BPR loss
__global__ void k_loss(const int* __restrict__ bd, int bi,
                       const float* __restrict__ total, float* __restrict__ acc) {
  int t    = blockIdx.x * blockDim.x + threadIdx.x;
  int row  = t >> 5;
  int lane = t & 31;
  if (row >= BATCHSZ) return;
  int u  = bd[row * 9 + bi * 3 + 0];
  int i0 = bd[row * 9 + bi * 3 + 1];
  int i1 = bd[row * 9 + bi * 3 + 2];
  float2 a  = ((const float2*)(total + (size_t)u * DD))[lane];
  float2 b0 = ((const float2*)(total + (size_t)(NU + i0) * DD))[lane];
  float2 b1 = ((const float2*)(total + (size_t)(NU + i1) * DD))[lane];
  float s0 = a.x * b0.x + a.y * b0.y;
  float s1 = a.x * b1.x + a.y * b1.y;
  #pragma unroll
  for (int m = 16; m > 0; m >>= 1) {
    s0 += __shfl_xor(s0, m, 32);
    s1 += __shfl_xor(s1, m, 32);
  }
  if (lane == 0) {
    float diff = s0 - s1;
    float sig  = 1.f / (1.f + expf(-diff));
    atomicAdd(acc, -logf(1e-10f + sig) * (1.f / (float)BATCHSZ));
  }
}

__global__ void k_final(const float* __restrict__ scal, float* __restrict__ out) {
  // scal[0]=sumsq(user_emb), scal[1]=sumsq(item_emb), scal[2]=BPR loss sum
  out[0] = scal[2] + 0.001f * (sqrtf(scal[0]) + sqrtf(scal[1])) / (float)NI;
}

// ---------------------------------------------------------------- launcher
extern "C" void kernel_launch(void* const* d_in, const int* in_sizes, int n_in,
                              void* d_out, int out_size, void* d_ws, size_t ws_size,
                              hipStream_t stream) {
  const int*   bd = (const int*)d_in[0];    // [4096,3,3]
  const int*   ei = (const int*)d_in[1];    // [3,2,E]
  const float* ue = (const float*)d_in[2];  // [50001,64]
  const float* ie = (const float*)d_in[3];  // [100001,64]
  const float* gw = (const float*)d_in[4];  // [3,2,64,64]
  const float* gb = (const float*)d_in[5];  // [3,2,64]
  const float* lw = (const float*)d_in[6];  // [3,NN,64]
  float* out = (float*)d_out;

  float* ws    = (float*)d_ws;
  float* total = ws;
  float* P     = total + (size_t)NN * DD;
  float* Q     = P     + (size_t)NN * DD;
  float* deg   = Q     + (size_t)NN * DD;
  float* dinv  = deg + NN;
  float* scal  = dinv + NN;                  // [0]=su [1]=si [2]=loss

  const size_t nTot = (size_t)NN * DD;
  const int bTot    = (int)((nTot + 255) / 256);
  const int bNodes  = (NN + 255) / 256;
  const int bEdges  = (NE + 255) / 256;
  const int strips  = (NN + 15) / 16;
  const int bGemm   = (strips + 7) / 8;      // 8 waves (256 threads) per block
  const int bScat   = (int)(((long long)NE * 32 + 255) / 256);
  const int bComb   = (int)(((long long)NN * 32 + 255) / 256);
  const int bLoss   = (BATCHSZ * 32 + 255) / 256;

  // init total + accumulators
  k_init_total<<<bTot, 256, 0, stream>>>(ue, ie, total);
  k_zero<<<1, 32, 0, stream>>>(scal, 3);
  k_sumsq<<<2048, 256, 0, stream>>>(ue, (size_t)NU * DD, &scal[0]);
  k_sumsq<<<2048, 256, 0, stream>>>(ie, (size_t)NI * DD, &scal[1]);

  for (int bi = 0; bi < 3; ++bi) {
    const int* src = ei + (size_t)bi * 2 * NE;
    const int* dst = src + NE;

    k_zero<<<bNodes, 256, 0, stream>>>(deg, NN);
    k_degree<<<bEdges, 256, 0, stream>>>(dst, deg);
    k_dinv<<<bNodes, 256, 0, stream>>>(deg, dinv);
    k_copy<<<bTot, 256, 0, stream>>>(total, P, nTot);

    for (int li = 0; li < 2; ++li) {
      const float* W = gw + ((size_t)bi * 2 + li) * DD * DD;
      const float* B = gb + ((size_t)bi * 2 + li) * DD;
      k_gemm_wmma<<<bGemm, 256, 0, stream>>>(P, W, Q, NN);
      k_bias_init<<<bTot, 256, 0, stream>>>(P, B);
      k_scatter<<<bScat, 256, 0, stream>>>(src, dst, dinv, Q, P);
    }

    k_combine<<<bComb, 256, 0, stream>>>(P, lw + (size_t)bi * NN * DD, total);
    k_loss<<<bLoss, 256, 0, stream>>>(bd, bi, total, &scal[2]);
  }

  k_final<<<1, 1, 0, stream>>>(scal, out);
}